// Bipartite_Layers_19310172963573
// MI455X (gfx1250) — compile-verified
//
#include <hip/hip_runtime.h>

typedef float v2f __attribute__((ext_vector_type(2)));
typedef float v8f __attribute__((ext_vector_type(8)));

constexpr int kB = 16, kT = 4096, kD = 64, kF = 128, kA = 16;
constexpr int kM = kB * kT;          // 65536 rows for the big GEMMs
constexpr int kAgg = 2 * kF * kA;    // 4096
constexpr int SPLIT = 8;             // T-splitting for pooling

// ---------------------------------------------------------------------------
// Generic fp32 WMMA GEMM: C[M,N] = A[M,K] @ B[K,N] + bias[N]
// Row-major, lda=K, ldb=N, ldc=N. One wave computes one 16x16 tile using
// V_WMMA_F32_16X16X4_F32 with a K-loop of step 4. Used for the small GEMMs
// (scores: N=16, aggregator: M=16/K=4096).
// ---------------------------------------------------------------------------
__global__ __launch_bounds__(256) void wmma_gemm16_bias(
    const float* __restrict__ A, const float* __restrict__ Bm,
    const float* __restrict__ bias, float* __restrict__ C,
    int M, int N, int K)
{
    int wave = (blockIdx.x * 256 + threadIdx.x) >> 5;
    int ntN = N >> 4;
    int totalTiles = (M >> 4) * ntN;
    if (wave >= totalTiles) return;            // wave-uniform: EXEC stays all-1s
    int mt = wave / ntN;
    int nt = wave - mt * ntN;
    int lane = threadIdx.x & 31;
    int l = lane & 15;                         // M (for A) / N (for B,C) within tile
    int h = lane >> 4;                         // K-half selector

    const float* arow = A + (mt * 16 + l) * K + 2 * h;       // A[m][k0+2h .. +1]
    const float* bcol = Bm + (2 * h) * N + nt * 16 + l;      // B[k0+2h][n]

    v8f acc = {0.f, 0.f, 0.f, 0.f, 0.f, 0.f, 0.f, 0.f};
    for (int k = 0; k < K; k += 4) {
        v2f a; a.x = arow[k];         a.y = arow[k + 1];
        v2f b; b.x = bcol[k * N];     b.y = bcol[k * N + N];
        acc = __builtin_amdgcn_wmma_f32_16x16x4_f32(
            false, a, false, b, (short)0, acc, false, false);
    }

    float bn = bias[nt * 16 + l];
    float* crow = C + (mt * 16 + 8 * h) * N + nt * 16 + l;   // row v + 8h
#pragma unroll
    for (int v = 0; v < 8; ++v) crow[v * N] = acc[v] + bn;
}

// ---------------------------------------------------------------------------
// Wide-tile fp32 WMMA GEMM for K=64, N=128: wave computes a 16x64 tile
// (4 N-subtiles share each A load -> 64 WMMAs per wave, A read once).
// Epilogue selected at compile time:
//   PERB=false: C = A@B + bias[n]                   (Xp = X@W_in + b_in)
//   PERB=true : C = A@B + bias[(row/T)*128 + n]     (out, fused agg bias)
//   RELU applied when requested.
// ---------------------------------------------------------------------------
template <bool RELU, bool PERB>
__global__ __launch_bounds__(256) void wmma_gemm_k64_n128(
    const float* __restrict__ A, const float* __restrict__ Bm,
    const float* __restrict__ bias, float* __restrict__ C)
{
    int wave = (blockIdx.x * 256 + threadIdx.x) >> 5;  // 0..8191
    int mt = wave >> 1;
    int n0 = (wave & 1) * 64;
    int lane = threadIdx.x & 31;
    int l = lane & 15;
    int h = lane >> 4;

    const float* arow = A + (mt * 16 + l) * kD + 2 * h;
    v8f acc[4] = {};
    for (int k = 0; k < kD; k += 4) {
        v2f a; a.x = arow[k]; a.y = arow[k + 1];
        const float* brow = Bm + (k + 2 * h) * kF + n0 + l;
#pragma unroll
        for (int j = 0; j < 4; ++j) {
            v2f b; b.x = brow[j * 16]; b.y = brow[j * 16 + kF];
            acc[j] = __builtin_amdgcn_wmma_f32_16x16x4_f32(
                false, a, false, b, (short)0, acc[j], false, false);
        }
    }

    int r0 = mt * 16 + 8 * h;
    int bidx = (mt * 16) >> 12;                 // row / T; a tile never crosses b
#pragma unroll
    for (int j = 0; j < 4; ++j) {
        int col = n0 + j * 16 + l;
        float bn = PERB ? bias[bidx * kF + col] : bias[col];
        float* crow = C + r0 * kF + col;
#pragma unroll
        for (int v = 0; v < 8; ++v) {
            float val = acc[j][v] + bn;
            crow[v * kF] = RELU ? (val > 0.f ? val : 0.f) : val;
        }
    }
}

// ---------------------------------------------------------------------------
// Softmax over the minibatch dim (axis 0). One thread per (t,a) flat index.
// S layout: [B, T*A] with stride kT*kA between batches.
// ---------------------------------------------------------------------------
__global__ __launch_bounds__(256) void softmax_over_b(
    const float* __restrict__ Sraw, float* __restrict__ S)
{
    int idx = blockIdx.x * 256 + threadIdx.x;       // t*kA + a, < 65536
    const int stride = kT * kA;
    float v[kB];
    float m = -3.402823466e38f;
#pragma unroll
    for (int b = 0; b < kB; ++b) { v[b] = Sraw[b * stride + idx]; m = fmaxf(m, v[b]); }
    float s = 0.f;
#pragma unroll
    for (int b = 0; b < kB; ++b) { v[b] = __expf(v[b] - m); s += v[b]; }
    float inv = 1.f / s;
#pragma unroll
    for (int b = 0; b < kB; ++b) S[b * stride + idx] = v[b] * inv;
}

// ---------------------------------------------------------------------------
// Streaming max/mean pooling over t, split SPLIT ways along T.
// Block = (b, a, s); thread f in [0,128). edges[b,a,f,t] = Xp[b,t,f]*S[b,t,a]
// never materialized.
// ---------------------------------------------------------------------------
__global__ __launch_bounds__(128) void pool_partial(
    const float* __restrict__ Xp, const float* __restrict__ S,
    float* __restrict__ pmax, float* __restrict__ psum)
{
    int s = blockIdx.x % SPLIT;
    int a = (blockIdx.x / SPLIT) % kA;
    int b = blockIdx.x / (SPLIT * kA);
    int f = threadIdx.x;
    const int chunk = kT / SPLIT;
    int t0 = s * chunk;

    float mx = -3.402823466e38f, sm = 0.f;
    for (int t = t0; t < t0 + chunk; ++t) {
        float sv = S[(b * kT + t) * kA + a];     // broadcast within block
        float xv = Xp[(b * kT + t) * kF + f];    // coalesced
        float p = xv * sv;
        mx = fmaxf(mx, p);
        sm += p;
    }
    int o = ((b * kA + a) * SPLIT + s) * kF + f;
    pmax[o] = mx;
    psum[o] = sm;
}

// Combine SPLIT partials; write aggregator [B, 2*F*A] straight into d_out tail.
__global__ __launch_bounds__(256) void pool_combine(
    const float* __restrict__ pmax, const float* __restrict__ psum,
    float* __restrict__ agg)
{
    int idx = blockIdx.x * 256 + threadIdx.x;    // < B*A*F = 32768
    int f = idx % kF;
    int a = (idx / kF) % kA;
    int b = idx / (kF * kA);
    float mx = -3.402823466e38f, sm = 0.f;
#pragma unroll
    for (int s = 0; s < SPLIT; ++s) {
        int o = ((b * kA + a) * SPLIT + s) * kF + f;
        mx = fmaxf(mx, pmax[o]);
        sm += psum[o];
    }
    agg[b * kAgg + a * (2 * kF) + f]      = mx;
    agg[b * kAgg + a * (2 * kF) + kF + f] = sm * (1.f / kT);
}

// ---------------------------------------------------------------------------
extern "C" void kernel_launch(void* const* d_in, const int* in_sizes, int n_in,
                              void* d_out, int out_size, void* d_ws, size_t ws_size,
                              hipStream_t stream) {
    (void)in_sizes; (void)n_in; (void)out_size; (void)ws_size;
    const float* X       = (const float*)d_in[0];   // [B,T,D]
    const float* W_in    = (const float*)d_in[1];   // [D,F]
    const float* b_in    = (const float*)d_in[2];   // [F]
    const float* W_score = (const float*)d_in[3];   // [D,A]
    const float* b_score = (const float*)d_in[4];   // [A]
    const float* W_out   = (const float*)d_in[5];   // [D+2FA, F]
    const float* b_out   = (const float*)d_in[6];   // [F]

    float* out = (float*)d_out;                     // [B,T,F]
    float* agg = out + (size_t)kM * kF;             // [B, 2*F*A] (output #2)

    float* ws   = (float*)d_ws;
    float* Xp   = ws;                         size_t off = (size_t)kM * kF;   // 32 MB
    float* Sraw = ws + off;                   off += (size_t)kM * kA;         // 4 MB
    float* S    = ws + off;                   off += (size_t)kM * kA;         // 4 MB
    float* pmax = ws + off;                   off += (size_t)kB * kA * SPLIT * kF;
    float* psum = ws + off;                   off += (size_t)kB * kA * SPLIT * kF;
    float* aggout = ws + off;                 // [B, F]

    // 1) Xp = X @ W_in + b_in : 8192 waves (16x64 tiles) -> 1024 blocks
    wmma_gemm_k64_n128<false, false><<<1024, 256, 0, stream>>>(X, W_in, b_in, Xp);
    // 2) Sraw = X @ W_score + b_score : 4096 waves (16x16 tiles) -> 512 blocks
    wmma_gemm16_bias<<<512, 256, 0, stream>>>(X, W_score, b_score, Sraw, kM, kA, kD);
    // 3) softmax over batch dim: 65536 threads
    softmax_over_b<<<256, 256, 0, stream>>>(Sraw, S);
    // 4) pooling partials: B*A*SPLIT = 2048 blocks x 128 threads
    pool_partial<<<kB * kA * SPLIT, 128, 0, stream>>>(Xp, S, pmax, psum);
    // 5) combine -> aggregator into d_out tail: 32768 threads
    pool_combine<<<128, 256, 0, stream>>>(pmax, psum, agg);
    // 6) agg_out = agg @ W_out[64:,:] + b_out : M=16,N=128,K=4096 -> 8 waves
    wmma_gemm16_bias<<<1, 256, 0, stream>>>(agg, W_out + (size_t)kD * kF, b_out,
                                            aggout, kB, kF, kAgg);
    // 7) out = relu(X @ W_out[:64,:] + agg_out[b]) : 8192 waves -> 1024 blocks
    wmma_gemm_k64_n128<true, true><<<1024, 256, 0, stream>>>(X, W_out, aggout, out);
}